// GMMVAE_1271310319877
// MI455X (gfx1250) — compile-verified
//
#include <hip/hip_runtime.h>
#include <hip/hip_bf16.h>

typedef float v2f __attribute__((ext_vector_type(2)));
typedef float v8f __attribute__((ext_vector_type(8)));

#define B_N    16384
#define X_DIM  784
#define Z_DIM  64
#define K_N    64
#define LOG2PI 1.8378770664093453f

// ---------------------------------------------------------------------------
// workspace layout (in floats)
//  acc[0]=SSE  acc[1]=kl_y  acc[2]=kl_z
#define ACC_OFF   ((size_t)0)
#define DK_OFF    ((size_t)64)
#define LK_OFF    ((size_t)128)
#define LQ_OFF    ((size_t)256)                       // B floats
#define BMAT_OFF  (LQ_OFF + (size_t)B_N)              // 64*128
#define Z_OFF     (BMAT_OFF + (size_t)64 * 128)       // B*64
#define Z2Z_OFF   (Z_OFF + (size_t)B_N * 64)          // B*128  = [z^2 | z]
#define UQ_OFF    (Z2Z_OFF + (size_t)B_N * 128)       // B*128  = [qs^2+qmu^2 | qmu]

// ---------------------------------------------------------------------------
__global__ void zero_acc_kernel(float* acc) {
    if (threadIdx.x < 4) acc[threadIdx.x] = 0.0f;
}

// per-b prep: z, z^2, qs^2+qmu^2, sum_z qz_logsigma     grid=B, block=64
__global__ void prep_b_kernel(const float* __restrict__ qmu,
                              const float* __restrict__ qls,
                              const float* __restrict__ eps,
                              float* __restrict__ Z,
                              float* __restrict__ Z2Z,
                              float* __restrict__ UQ,
                              float* __restrict__ Lq) {
    const int b = blockIdx.x;
    const int z = threadIdx.x;
    const size_t i = (size_t)b * Z_DIM + z;
    const float ls = qls[i];
    const float qs = __expf(ls);
    const float mu = qmu[i];
    const float zz = mu + qs * eps[i];
    Z[i] = zz;
    Z2Z[(size_t)b * 128 + z]      = zz * zz;
    Z2Z[(size_t)b * 128 + 64 + z] = zz;
    UQ[(size_t)b * 128 + z]       = qs * qs + mu * mu;
    UQ[(size_t)b * 128 + 64 + z]  = mu;

    float v = ls;
    for (int off = 16; off > 0; off >>= 1) v += __shfl_down(v, off, 32);
    __shared__ float part[2];
    if ((threadIdx.x & 31) == 0) part[threadIdx.x >> 5] = v;
    __syncthreads();
    if (threadIdx.x == 0) Lq[b] = part[0] + part[1];
}

// per-k prep: Bmat[k] = [R | -2*mu*R],  Dk = sum mu^2 R,  Lk = sum logsigma_p
// grid=K, block=64
__global__ void prep_k_kernel(const float* __restrict__ mu,
                              const float* __restrict__ lsp,
                              float* __restrict__ Bm,
                              float* __restrict__ Dk,
                              float* __restrict__ Lk) {
    const int k = blockIdx.x;
    const int z = threadIdx.x;
    const size_t i = (size_t)k * Z_DIM + z;
    const float ls = lsp[i];
    const float R  = __expf(-2.0f * ls);
    const float m  = mu[i];
    Bm[(size_t)k * 128 + z]      = R;
    Bm[(size_t)k * 128 + 64 + z] = -2.0f * m * R;

    float d = m * m * R, l = ls;
    for (int off = 16; off > 0; off >>= 1) {
        d += __shfl_down(d, off, 32);
        l += __shfl_down(l, off, 32);
    }
    __shared__ float pd[2], pl[2];
    if ((threadIdx.x & 31) == 0) { pd[threadIdx.x >> 5] = d; pl[threadIdx.x >> 5] = l; }
    __syncthreads();
    if (threadIdx.x == 0) { Dk[k] = pd[0] + pd[1]; Lk[k] = pl[0] + pl[1]; }
}

// ---------------------------------------------------------------------------
// Decoder GEMM + fused SSE: mean = Z @ W_dec + b_dec ; SSE += (x-mean)^2
// grid = B/16 row-tiles, block = 256 (8 waves). Each wave owns N-tiles
// wave, wave+8, ... (49 N-tiles of 16 cols). f32 WMMA 16x16x4, K=64 -> 16 steps.
__global__ __launch_bounds__(256) void decoder_sse_kernel(
        const float* __restrict__ Z, const float* __restrict__ W,
        const float* __restrict__ bdec, const float* __restrict__ x,
        float* __restrict__ acc) {
    const int row0 = blockIdx.x * 16;
    const int lane = threadIdx.x & 31;
    const int wave = threadIdx.x >> 5;
    const int lr = lane & 15;
    const int kh = lane >> 4;               // K-pair half (A/B layout, ISA 7.12.2)
    const int aRow = row0 + lr;

    // hoist A (16 k-steps x 2 floats) into registers; reused for every N tile
    v2f a[16];
#pragma unroll
    for (int s = 0; s < 16; ++s)
        a[s] = *(const v2f*)(Z + (size_t)aRow * Z_DIM + 4 * s + 2 * kh);

    float sse = 0.0f;
    for (int nt = wave; nt < 49; nt += 8) {
        const int col = nt * 16 + lr;
        v8f c = {};
#pragma unroll
        for (int s = 0; s < 16; ++s) {
            const int kz = 4 * s + 2 * kh;
            v2f b;
            b.x = W[(size_t)kz * X_DIM + col];
            b.y = W[(size_t)(kz + 1) * X_DIM + col];
            c = __builtin_amdgcn_wmma_f32_16x16x4_f32(
                    false, a[s], false, b, (short)0, c, false, false);
        }
        const float bd = bdec[col];
#pragma unroll
        for (int j = 0; j < 8; ++j) {
            const int row = row0 + j + 8 * kh;      // C layout: vgpr j -> M=j / j+8
            const float m = c[j] + bd;
            const float d = x[(size_t)row * X_DIM + col] - m;
            sse += d * d;
        }
    }
    for (int off = 16; off > 0; off >>= 1) sse += __shfl_down(sse, off, 32);
    if (lane == 0) atomicAdd(&acc[0], sse);
}

// ---------------------------------------------------------------------------
// Mixture kernel: Q1 = [z^2|z]@Bm^T, Q2 = [qs^2+qmu^2|qmu]@Bm^T  (K=128, WMMA),
// then per-row softmax over k, kl_y and kl_z accumulation.
// grid = B/16, block = 256: wave = sel*4 + n_tile (2 matmuls x 4 col-tiles).
__global__ __launch_bounds__(256) void mixture_kernel(
        const float* __restrict__ A1, const float* __restrict__ A2,
        const float* __restrict__ Bm, const float* __restrict__ Dk,
        const float* __restrict__ Lk, const float* __restrict__ Lq,
        const float* __restrict__ pi, float* __restrict__ acc) {
    __shared__ float Q[2][16][64];
    __shared__ float pzyS[16][64];
    __shared__ float lpiS[16][64];
    __shared__ float klzS[16][64];

    const int row0 = blockIdx.x * 16;
    const int lane = threadIdx.x & 31;
    const int wave = threadIdx.x >> 5;
    const int sel = wave >> 2;
    const int nt  = wave & 3;
    const int lr = lane & 15;
    const int kh = lane >> 4;
    const float* A = sel ? A2 : A1;
    const size_t aBase = (size_t)(row0 + lr) * 128;
    const size_t bBase = (size_t)(nt * 16 + lr) * 128;

    v8f c = {};
#pragma unroll
    for (int s = 0; s < 32; ++s) {
        const int kz = 4 * s + 2 * kh;
        v2f a = *(const v2f*)(A + aBase + kz);
        v2f b = *(const v2f*)(Bm + bBase + kz);
        c = __builtin_amdgcn_wmma_f32_16x16x4_f32(
                false, a, false, b, (short)0, c, false, false);
    }
#pragma unroll
    for (int j = 0; j < 8; ++j)
        Q[sel][j + 8 * kh][nt * 16 + lr] = c[j];
    __syncthreads();

    if (threadIdx.x < 16) {
        const int r = threadIdx.x;
        const int b = row0 + r;
        const float lqb = Lq[b];
        float mx = -1e30f;
        for (int k = 0; k < K_N; ++k) {
            const float lpi = __logf(pi[(size_t)b * K_N + k]);
            const float comp = -0.5f * (Q[0][r][k] + Dk[k]) - Lk[k] - 32.0f * LOG2PI;
            const float pzy = comp + lpi;
            pzyS[r][k] = pzy;
            lpiS[r][k] = lpi;
            klzS[r][k] = Lk[k] - lqb + 0.5f * (Q[1][r][k] + Dk[k]) - 32.0f;
            mx = fmaxf(mx, pzy);
        }
        float se = 0.0f;
        for (int k = 0; k < K_N; ++k) se += __expf(pzyS[r][k] - mx);
        const float lse = mx + __logf(se);
        float kly = 0.0f, klz = 0.0f;
        for (int k = 0; k < K_N; ++k) {
            const float lpy = pzyS[r][k] - lse;
            const float p = __expf(lpy);
            kly += p * (lpy - lpiS[r][k]);
            klz += p * klzS[r][k];
        }
        atomicAdd(&acc[1], kly);
        atomicAdd(&acc[2], klz);
    }
}

__global__ void final_kernel(const float* __restrict__ acc, float* __restrict__ out) {
    const float log_px_z = -0.5f * acc[0]
                         - 0.5f * LOG2PI * (float)((size_t)B_N * X_DIM);
    out[0] = (log_px_z - acc[1] - acc[2]) / (float)B_N;
}

// ---------------------------------------------------------------------------
extern "C" void kernel_launch(void* const* d_in, const int* in_sizes, int n_in,
                              void* d_out, int out_size, void* d_ws, size_t ws_size,
                              hipStream_t stream) {
    const float* x    = (const float*)d_in[0];
    const float* pi   = (const float*)d_in[1];
    const float* qmu  = (const float*)d_in[2];
    const float* qls  = (const float*)d_in[3];
    const float* eps  = (const float*)d_in[4];
    const float* pmu  = (const float*)d_in[5];
    const float* plsp = (const float*)d_in[6];
    const float* Wd   = (const float*)d_in[7];
    const float* bd   = (const float*)d_in[8];
    float* out = (float*)d_out;

    float* ws   = (float*)d_ws;
    float* acc  = ws + ACC_OFF;
    float* Dk   = ws + DK_OFF;
    float* Lk   = ws + LK_OFF;
    float* Lq   = ws + LQ_OFF;
    float* Bm   = ws + BMAT_OFF;
    float* Z    = ws + Z_OFF;
    float* Z2Z  = ws + Z2Z_OFF;
    float* UQ   = ws + UQ_OFF;

    zero_acc_kernel<<<1, 32, 0, stream>>>(acc);
    prep_b_kernel<<<B_N, Z_DIM, 0, stream>>>(qmu, qls, eps, Z, Z2Z, UQ, Lq);
    prep_k_kernel<<<K_N, Z_DIM, 0, stream>>>(pmu, plsp, Bm, Dk, Lk);
    decoder_sse_kernel<<<B_N / 16, 256, 0, stream>>>(Z, Wd, bd, x, acc);
    mixture_kernel<<<B_N / 16, 256, 0, stream>>>(Z2Z, UQ, Bm, Dk, Lk, Lq, pi, acc);
    final_kernel<<<1, 1, 0, stream>>>(acc, out);
}